// BayesianFlowNetworkDiscretised_51548197486997
// MI455X (gfx1250) — compile-verified
//
#include <hip/hip_runtime.h>
#include <math.h>

// MI455X / gfx1250. wave32. One lane owns one (b,d) element for the erf tail
// and the coalesced 64B store; the tiny 2->16->2 MLP runs on
// V_WMMA_F32_16X16X4_F32 in transposed form (rows = hidden units, cols = elems).
//
// Roofline: ~100.7 MB out + ~6.3 MB in => ~4.6us at 23.3 TB/s. VALU side is
// 15 erf + 16 gelu per element; all transcendentals are branch-free
// v_exp_f32 / v_rcp_f32 / v_sqrt_f32 sequences to stay under the HBM bound.

typedef __attribute__((ext_vector_type(2))) float v2f;
typedef __attribute__((ext_vector_type(8))) float v8f;

#define SWZ_SWAPX16 0x401F  // ds_swizzle group-of-32: xor=0x10, or=0, and=0x1f

__device__ __forceinline__ float swz16(float x) {
  return __int_as_float(__builtin_amdgcn_ds_swizzle(__float_as_int(x), SWZ_SWAPX16));
}

// 0.5*(1+tanh(u)) == sigmoid(2u); gelu(x) = x * sigmoid(2u(x)).
// Branch-free: v_exp_f32 saturates to 0/inf, rcp handles both ends.
__device__ __forceinline__ float gelu_fast(float x) {
  float u = 0.7978845608028654f * __builtin_fmaf(0.044715f * x * x, x, x);
  float e = __builtin_amdgcn_exp2f(-2.8853900817779268f * u);  // exp(-2u)
  return x * __builtin_amdgcn_rcpf(1.0f + e);
}

// Abramowitz-Stegun 7.1.26, branch-free, |err| < 1.5e-7.
__device__ __forceinline__ float erf_fast(float x) {
  float a = __builtin_fabsf(x);
  float tt = __builtin_amdgcn_rcpf(__builtin_fmaf(0.3275911f, a, 1.0f));
  float p = __builtin_fmaf(tt, 1.061405429f, -1.453152027f);
  p = __builtin_fmaf(tt, p, 1.421413741f);
  p = __builtin_fmaf(tt, p, -0.284496736f);
  p = __builtin_fmaf(tt, p, 0.254829592f);
  p = p * tt;
  float e = __builtin_amdgcn_exp2f(-1.4426950408889634f * a * a);  // exp(-a^2)
  float r = __builtin_fmaf(-p, e, 1.0f);
  return __builtin_copysignf(r, x);
}

__global__ __launch_bounds__(256) void bfn_discretised_kernel(
    const float* __restrict__ mu, const float* __restrict__ tptr,
    const float* __restrict__ W1, const float* __restrict__ b1,
    const float* __restrict__ W2, const float* __restrict__ b2,
    float* __restrict__ out, int D) {
  const int lane = threadIdx.x & 31;
  const int wave = threadIdx.x >> 5;
  const int b    = blockIdx.y;
  const int d    = blockIdx.x * 256 + wave * 32 + lane;  // element this lane owns
  const bool lo  = lane < 16;
  const int  n   = lane & 15;

  const float t = tptr[b];

  // Clamped load so partial tiles keep EXEC all-ones through the WMMAs;
  // only the final store is predicated.
  const int dl = (d < D) ? d : (D - 1);
  const float mu_e  = mu[(size_t)b * D + dl];
  const float mu_sw = swz16(mu_e);  // lanes 0-15 now see elements 16-31's mu

  // ---- A1 = W1^T padded to 16x4 (rows n = hidden unit; K0=mu-w, K1=t-w) ----
  // Unconditional in-range loads, then cndmask (avoid exec-diverging loads).
  const float w1a = W1[n];
  const float w1b = W1[16 + n];
  v2f a1;
  a1.x = lo ? w1a : 0.0f;  // lanes 0-15 supply K=0,1; hi lanes K=2,3 -> 0
  a1.y = lo ? w1b : 0.0f;

  // ---- C1[n,m] = b1[n] broadcast along columns ----
  v8f c1;
#pragma unroll
  for (int v = 0; v < 8; ++v) c1[v] = lo ? b1[v] : b1[v + 8];

  // ---- C2[j,m] = b2[j] (rows j=0,1 only; j = v + 8*(lane>=16)) ----
  v8f c2base;
#pragma unroll
  for (int v = 0; v < 8; ++v) c2base[v] = 0.0f;
  c2base[0] = lo ? b2[0] : 0.0f;
  c2base[1] = lo ? b2[1] : 0.0f;

  // ---- A2 chunks = W2^T (2x16, padded), K split into 4 chunks of 4 ----
  v2f a2[4];
  const bool jv = (n < 2);
  const int  jj = n & 1;  // in-range column even for masked lanes
#pragma unroll
  for (int c = 0; c < 4; ++c) {
    int k0 = 4 * c + ((lane >> 4) << 1);  // hi half supplies K rows +2,+3
    float wx = W2[2 * (k0 + 0) + jj];
    float wy = W2[2 * (k0 + 1) + jj];
    a2[c].x = jv ? wx : 0.0f;
    a2[c].y = jv ? wy : 0.0f;
  }

  float mu_res[2], ls_res[2];  // per-tile (mu_eps, ln_sigma) in lanes 0-15
#pragma unroll
  for (int tile = 0; tile < 2; ++tile) {
    // B1: columns m = elements; row0 = mu_m, row1 = t, rows 2-3 = 0
    float muv = (tile == 0) ? mu_e : mu_sw;
    v2f b1m;
    b1m.x = lo ? muv : 0.0f;
    b1m.y = lo ? t   : 0.0f;

    // Layer 1: D1[n,m] = W1[0,n]*mu_m + W1[1,n]*t + b1[n]
    v8f h = __builtin_amdgcn_wmma_f32_16x16x4_f32(
        false, a1, false, b1m, (short)0, c1, false, false);

    // gelu elementwise; also make half-wave-swapped copies for B repacking
    float g[8], s[8];
#pragma unroll
    for (int v = 0; v < 8; ++v) g[v] = gelu_fast(h[v]);
#pragma unroll
    for (int v = 0; v < 8; ++v) s[v] = swz16(g[v]);

    // Repack G (D-layout) into B-operand layout per K-chunk.
    // D-layout: G[n,m] at lane = m + 16*(n/8), VGPR = n%8
    // B-layout chunk c, VGPR g: need G[4c + 2*(lane/16) + g, lane%16]
    v2f bm0, bm1, bm2, bm3;
    bm0.x = lo ? g[0] : s[2];  bm0.y = lo ? g[1] : s[3];
    bm1.x = lo ? g[4] : s[6];  bm1.y = lo ? g[5] : s[7];
    bm2.x = lo ? s[0] : g[2];  bm2.y = lo ? s[1] : g[3];
    bm3.x = lo ? s[4] : g[6];  bm3.y = lo ? s[5] : g[7];

    // Layer 2: D2[j,m] = sum_n W2[n,j]*G[n,m] + b2[j], K=16 via 4 chained WMMAs
    v8f acc = c2base;
    acc = __builtin_amdgcn_wmma_f32_16x16x4_f32(false, a2[0], false, bm0, (short)0, acc, false, false);
    acc = __builtin_amdgcn_wmma_f32_16x16x4_f32(false, a2[1], false, bm1, (short)0, acc, false, false);
    acc = __builtin_amdgcn_wmma_f32_16x16x4_f32(false, a2[2], false, bm2, (short)0, acc, false, false);
    acc = __builtin_amdgcn_wmma_f32_16x16x4_f32(false, a2[3], false, bm3, (short)0, acc, false, false);

    mu_res[tile] = acc[0];  // j=0 row lives in VGPR0, lanes 0-15
    ls_res[tile] = acc[1];  // j=1 row lives in VGPR1, lanes 0-15
  }

  // Redistribute: lane L owns element L. Tile0 -> lanes 0-15 local,
  // tile1 -> values sit in lanes 0-15, move to 16-31 via SWAPX16.
  const float mu_eps = lo ? mu_res[0] : swz16(mu_res[1]);
  const float ln_sig = lo ? ls_res[0] : swz16(ls_res[1]);

  // ---- per-element tail (all branch-free hardware trans ops) ----
  const float s2t = __builtin_amdgcn_exp2f(2.0f * t * -5.643856189774724f); // 0.02^(2t)
  const float gamma = 1.0f - s2t;
  const float rgam  = __builtin_amdgcn_rcpf(gamma);
  const float vscl  = __builtin_amdgcn_sqrtf(s2t * rgam);
  float mu_x = __builtin_fmaf(-vscl, mu_eps, mu_e * rgam);
  float lc = __builtin_fminf(__builtin_fmaxf(ln_sig, -10.0f), 10.0f);
  float sigma_x =
      __builtin_fmaxf(vscl * __builtin_amdgcn_exp2f(1.4426950408889634f * lc), 0.02f);
  const bool tiny = (t < 1e-6f);
  mu_x    = tiny ? 0.0f : mu_x;
  sigma_x = tiny ? 1.0f : sigma_x;
  const float inv  = 0.7071067811865476f * __builtin_amdgcn_rcpf(sigma_x);
  const float step = 0.125f * inv;          // boundary spacing * inv
  const float base = (-1.0f - mu_x) * inv;  // boundary 0 argument

  // 17 boundaries j/8-1; boundary 0 -> cdf 0, boundary 16 -> cdf 1.
  float r[16];
  float cdf_prev = 0.0f;
#pragma unroll
  for (int jb = 1; jb <= 16; ++jb) {
    float cdf = (jb == 16)
        ? 1.0f
        : __builtin_fmaf(0.5f, erf_fast(__builtin_fmaf((float)jb, step, base)), 0.5f);
    r[jb - 1] = cdf - cdf_prev;
    cdf_prev = cdf;
  }

  if (d < D) {
    float4* ov = (float4*)(out + (((size_t)b * D + d) << 4));
#pragma unroll
    for (int q = 0; q < 4; ++q)
      ov[q] = make_float4(r[4 * q], r[4 * q + 1], r[4 * q + 2], r[4 * q + 3]);
  }
}

extern "C" void kernel_launch(void* const* d_in, const int* in_sizes, int n_in,
                              void* d_out, int out_size, void* d_ws, size_t ws_size,
                              hipStream_t stream) {
  const float* mu = (const float*)d_in[0];
  const float* t  = (const float*)d_in[1];
  const float* W1 = (const float*)d_in[2];
  const float* b1 = (const float*)d_in[3];
  const float* W2 = (const float*)d_in[4];
  const float* b2 = (const float*)d_in[5];
  float* out = (float*)d_out;

  const int B = in_sizes[1];            // t is [B,1]
  const int D = in_sizes[0] / B;        // mu is [B,D]  (D = 49152)

  dim3 block(256, 1, 1);                // 8 waves * 32 elements
  dim3 grid((D + 255) / 256, B, 1);     // 192 x 32 for the ref shapes
  bfn_discretised_kernel<<<grid, block, 0, stream>>>(mu, t, W1, b1, W2, b2, out, D);
}